// WA_GCN_22625887715769
// MI455X (gfx1250) — compile-verified
//
#include <hip/hip_runtime.h>
#include <hip/hip_bf16.h>

#define N_NODES 100000
#define N_EDGES 3200000
#define N_GRAPH 128
#define F_IN    512
#define NC      128
#define HID     256
#define OUTC    10
#define SCAN_CHUNK 1024

typedef float v2f __attribute__((ext_vector_type(2)));
typedef float v8f __attribute__((ext_vector_type(8)));

// ---------------------------------------------------------------------------
// Integer degree counts (once): out-degree for norm_src, in-degree for
// norm_dst and for the CSR row offsets.
// ---------------------------------------------------------------------------
__global__ __launch_bounds__(256)
void gcn_degcnt(const int* __restrict__ src, const int* __restrict__ dst,
                int* __restrict__ outc, int* __restrict__ inc, int E) {
    int i = blockIdx.x * blockDim.x + threadIdx.x;
    if (i < E) {
        atomicAdd(&outc[src[i]], 1);
        atomicAdd(&inc[dst[i]], 1);
    }
}

__global__ __launch_bounds__(256)
void gcn_norms(const int* __restrict__ outc, const int* __restrict__ inc,
               float* __restrict__ ns, float* __restrict__ nd, int n) {
    int i = blockIdx.x * blockDim.x + threadIdx.x;
    if (i < n) {
        int o = outc[i]; ns[i] = (o > 0) ? rsqrtf((float)o) : 0.0f;
        int d = inc[i];  nd[i] = (d > 0) ? rsqrtf((float)d) : 0.0f;
    }
}

// ---------------------------------------------------------------------------
// Exclusive prefix scan of in-degree -> row_start  (3 tiny kernels, N=100K)
// ---------------------------------------------------------------------------
__global__ __launch_bounds__(256)
void gcn_scan1(const int* __restrict__ cnt, int* __restrict__ excl,
               int* __restrict__ blockSums, int n) {
    __shared__ int lds[256];
    const int t  = threadIdx.x;
    const int i0 = blockIdx.x * SCAN_CHUNK + t * 4;
    int v0 = (i0 + 0 < n) ? cnt[i0 + 0] : 0;
    int v1 = (i0 + 1 < n) ? cnt[i0 + 1] : 0;
    int v2 = (i0 + 2 < n) ? cnt[i0 + 2] : 0;
    int v3 = (i0 + 3 < n) ? cnt[i0 + 3] : 0;
    const int tsum = v0 + v1 + v2 + v3;
    lds[t] = tsum;
    __syncthreads();
    for (int off = 1; off < 256; off <<= 1) {       // Hillis-Steele inclusive
        int x = (t >= off) ? lds[t - off] : 0;
        __syncthreads();
        lds[t] += x;
        __syncthreads();
    }
    if (t == 255) blockSums[blockIdx.x] = lds[255];
    int run = lds[t] - tsum;                         // exclusive thread prefix
    if (i0 + 0 < n) { excl[i0 + 0] = run; } run += v0;
    if (i0 + 1 < n) { excl[i0 + 1] = run; } run += v1;
    if (i0 + 2 < n) { excl[i0 + 2] = run; } run += v2;
    if (i0 + 3 < n) { excl[i0 + 3] = run; }
}

__global__ void gcn_scan2(int* __restrict__ blockSums, int nb) {
    if (blockIdx.x == 0 && threadIdx.x == 0) {
        int run = 0;
        for (int b = 0; b < nb; ++b) { int v = blockSums[b]; blockSums[b] = run; run += v; }
    }
}

__global__ __launch_bounds__(256)
void gcn_scan3(int* __restrict__ excl, const int* __restrict__ blockSums, int n, int E) {
    int i = blockIdx.x * blockDim.x + threadIdx.x;
    if (i < n) excl[i] += blockSums[i / SCAN_CHUNK];
    if (i == 0) excl[n] = E;                         // sentinel: row_start[N] = E
}

// Bucket fill: csr_src[p] = src of each in-edge, grouped by dst.
__global__ __launch_bounds__(256)
void gcn_csr_fill(const int* __restrict__ src, const int* __restrict__ dst,
                  int* __restrict__ cursor, int* __restrict__ csr_src, int E) {
    int e = blockIdx.x * blockDim.x + threadIdx.x;
    if (e < E) {
        int p = atomicAdd(&cursor[dst[e]], 1);
        csr_src[p] = src[e];
    }
}

// ---------------------------------------------------------------------------
// CSR aggregation: agg[n] = sum over in-edges of hs[src]  (hs pre-scaled by
// norm_src). One wave per node; lanes hold a float4 (d=128) or 2xfloat4
// (d=256) slice in registers. Unrolled 4 edges deep: the 4 index loads and
// all row loads are independent, so 4+ gathers stay in flight per wave
// (this loop is the roofline-dominant phase; it is latency-bound otherwise).
// ---------------------------------------------------------------------------
__global__ __launch_bounds__(256)
void gcn_csr_agg(const float* __restrict__ hs, const int* __restrict__ row_start,
                 const int* __restrict__ csr_src, float* __restrict__ agg,
                 int Nn, int d) {
    const int lane = threadIdx.x & 31;
    const int node = (blockIdx.x * blockDim.x + threadIdx.x) >> 5;
    if (node >= Nn) return;                          // wave-uniform
    const int p0 = row_start[node];
    const int p1 = row_start[node + 1];
    const int c  = lane * 4;
    if (d == 128) {
        float4 a = {0.f, 0.f, 0.f, 0.f};
        int p = p0;
        for (; p + 4 <= p1; p += 4) {
            const int s0 = csr_src[p + 0];
            const int s1 = csr_src[p + 1];
            const int s2 = csr_src[p + 2];
            const int s3 = csr_src[p + 3];
            float4 w0 = *(const float4*)(hs + (size_t)s0 * 128 + c);
            float4 w1 = *(const float4*)(hs + (size_t)s1 * 128 + c);
            float4 w2 = *(const float4*)(hs + (size_t)s2 * 128 + c);
            float4 w3 = *(const float4*)(hs + (size_t)s3 * 128 + c);
            a.x += w0.x; a.y += w0.y; a.z += w0.z; a.w += w0.w;
            a.x += w1.x; a.y += w1.y; a.z += w1.z; a.w += w1.w;
            a.x += w2.x; a.y += w2.y; a.z += w2.z; a.w += w2.w;
            a.x += w3.x; a.y += w3.y; a.z += w3.z; a.w += w3.w;
        }
        for (; p < p1; ++p) {
            const int s = csr_src[p];
            float4 v = *(const float4*)(hs + (size_t)s * 128 + c);
            a.x += v.x; a.y += v.y; a.z += v.z; a.w += v.w;
        }
        *(float4*)(agg + (size_t)node * 128 + c) = a;
    } else {
        float4 a0 = {0.f, 0.f, 0.f, 0.f};
        float4 a1 = {0.f, 0.f, 0.f, 0.f};
        int p = p0;
        for (; p + 4 <= p1; p += 4) {
            const int s0 = csr_src[p + 0];
            const int s1 = csr_src[p + 1];
            const int s2 = csr_src[p + 2];
            const int s3 = csr_src[p + 3];
            const float* __restrict__ h0 = hs + (size_t)s0 * 256;
            const float* __restrict__ h1 = hs + (size_t)s1 * 256;
            const float* __restrict__ h2 = hs + (size_t)s2 * 256;
            const float* __restrict__ h3 = hs + (size_t)s3 * 256;
            float4 u0 = *(const float4*)(h0 + c);
            float4 w0 = *(const float4*)(h0 + c + 128);
            float4 u1 = *(const float4*)(h1 + c);
            float4 w1 = *(const float4*)(h1 + c + 128);
            float4 u2 = *(const float4*)(h2 + c);
            float4 w2 = *(const float4*)(h2 + c + 128);
            float4 u3 = *(const float4*)(h3 + c);
            float4 w3 = *(const float4*)(h3 + c + 128);
            a0.x += u0.x; a0.y += u0.y; a0.z += u0.z; a0.w += u0.w;
            a1.x += w0.x; a1.y += w0.y; a1.z += w0.z; a1.w += w0.w;
            a0.x += u1.x; a0.y += u1.y; a0.z += u1.z; a0.w += u1.w;
            a1.x += w1.x; a1.y += w1.y; a1.z += w1.z; a1.w += w1.w;
            a0.x += u2.x; a0.y += u2.y; a0.z += u2.z; a0.w += u2.w;
            a1.x += w2.x; a1.y += w2.y; a1.z += w2.z; a1.w += w2.w;
            a0.x += u3.x; a0.y += u3.y; a0.z += u3.z; a0.w += u3.w;
            a1.x += w3.x; a1.y += w3.y; a1.z += w3.z; a1.w += w3.w;
        }
        for (; p < p1; ++p) {
            const int s = csr_src[p];
            const float* __restrict__ hp = hs + (size_t)s * 256;
            float4 v0 = *(const float4*)(hp + c);
            float4 v1 = *(const float4*)(hp + c + 128);
            a0.x += v0.x; a0.y += v0.y; a0.z += v0.z; a0.w += v0.w;
            a1.x += v1.x; a1.y += v1.y; a1.z += v1.z; a1.w += v1.w;
        }
        *(float4*)(agg + (size_t)node * 256 + c)       = a0;
        *(float4*)(agg + (size_t)node * 256 + c + 128) = a1;
    }
}

// ---------------------------------------------------------------------------
// fp32 WMMA GEMM: C[M,Nc] = rowScale(A)[M,K] * B + bias
//   transB=1: B given as [Nc,K] row-major;  transB=0: B given as [K,Nc].
// One wave per 16x16 C tile, V_WMMA_F32_16X16X4_F32, K step 4.
// ---------------------------------------------------------------------------
__global__ __launch_bounds__(256)
void gcn_gemm_wmma_f32(const float* __restrict__ A, const float* __restrict__ B,
                       float* __restrict__ C,
                       const float* __restrict__ rowScale,   // nullable
                       const float* __restrict__ bias,       // nullable
                       int M, int K, int Nc, int transB) {
    const int lane = threadIdx.x & 31;
    const int wave = threadIdx.x >> 5;
    const int half = lane >> 4;
    const int r    = lane & 15;
    const int ntn  = Nc >> 4;
    const int ntiles = (M >> 4) * ntn;
    int tile = blockIdx.x * (blockDim.x >> 5) + wave;
    if (tile >= ntiles) return;                      // wave-uniform
    const int tm = tile / ntn;
    const int tn = tile - tm * ntn;
    const int arow = (tm << 4) + r;
    const int col  = (tn << 4) + r;

    const float scale = rowScale ? rowScale[arow] : 1.0f;
    const float* __restrict__ ap = A + (size_t)arow * K + 2 * half;
    v8f acc = {};
    if (transB) {
        const float* __restrict__ bp = B + (size_t)col * K + 2 * half;
        for (int k = 0; k < K; k += 4) {
            v2f a = *(const v2f*)(ap + k);
            a.x *= scale; a.y *= scale;
            v2f b = *(const v2f*)(bp + k);
            acc = __builtin_amdgcn_wmma_f32_16x16x4_f32(
                false, a, false, b, (short)0, acc, false, false);
        }
    } else {
        for (int k = 0; k < K; k += 4) {
            v2f a = *(const v2f*)(ap + k);
            a.x *= scale; a.y *= scale;
            v2f b;
            b.x = B[(size_t)(k + 2 * half)     * Nc + col];
            b.y = B[(size_t)(k + 2 * half + 1) * Nc + col];
            acc = __builtin_amdgcn_wmma_f32_16x16x4_f32(
                false, a, false, b, (short)0, acc, false, false);
        }
    }
    const float bv = bias ? bias[col] : 0.0f;
#pragma unroll
    for (int v = 0; v < 8; ++v) {
        C[(size_t)((tm << 4) + v + (half << 3)) * Nc + col] = acc[v] + bv;
    }
}

// ---------------------------------------------------------------------------
// BN batch stats over [M, 256]
// ---------------------------------------------------------------------------
__global__ __launch_bounds__(256)
void gcn_colstats(const float* __restrict__ h, float* __restrict__ sums,
                  float* __restrict__ sumsq, int M, int rowsPerBlock) {
    const int c  = threadIdx.x;
    const int r0 = blockIdx.x * rowsPerBlock;
    int r1 = r0 + rowsPerBlock; if (r1 > M) r1 = M;
    float s = 0.0f, q = 0.0f;
    for (int r = r0; r < r1; ++r) {
        float v = h[(size_t)r * HID + c];
        s += v; q += v * v;
    }
    atomicAdd(&sums[c], s);
    atomicAdd(&sumsq[c], q);
}

// BN (biased var) + ReLU, optional post-scale by norm_src (folds the next
// layer's edge scale so the gather loop is pure adds).
__global__ __launch_bounds__(256)
void gcn_bn_relu(float* __restrict__ h, const float* __restrict__ sums,
                 const float* __restrict__ sumsq, const float* __restrict__ gamma,
                 const float* __restrict__ beta, const float* __restrict__ postScale,
                 int M) {
    size_t i = (size_t)blockIdx.x * blockDim.x + threadIdx.x;
    if (i >= (size_t)M * HID) return;
    const int c = (int)(i & (HID - 1));
    const float inv_m = 1.0f / (float)M;
    const float mu  = sums[c] * inv_m;
    const float var = sumsq[c] * inv_m - mu * mu;
    float v = (h[i] - mu) * rsqrtf(var + 1e-5f) * gamma[c] + beta[c];
    v = v > 0.0f ? v : 0.0f;
    if (postScale) v *= postScale[i >> 8];           // node index (HID==256)
    h[i] = v;
}

// ---------------------------------------------------------------------------
// Per-graph max pool via integer atomicMax on non-negative float bits.
// ---------------------------------------------------------------------------
__global__ __launch_bounds__(256)
void gcn_pool_max(const float* __restrict__ h, const int* __restrict__ gid,
                  unsigned* __restrict__ hg, int M) {
    size_t i = (size_t)blockIdx.x * blockDim.x + threadIdx.x;
    if (i >= (size_t)M * HID) return;
    const int n = (int)(i >> 8);
    const int c = (int)(i & (HID - 1));
    atomicMax(&hg[(size_t)gid[n] * HID + c], __float_as_uint(h[i]));
}

__global__ __launch_bounds__(256)
void gcn_fc(const float* __restrict__ hg, const float* __restrict__ w,
            const float* __restrict__ b, float* __restrict__ out) {
    int i = blockIdx.x * blockDim.x + threadIdx.x;
    if (i >= N_GRAPH * OUTC) return;
    const int g = i / OUTC, o = i - g * OUTC;
    float acc = b[o];
    const float* __restrict__ hr = hg + (size_t)g * HID;
    for (int c = 0; c < HID; ++c) acc += hr[c] * w[c * OUTC + o];
    out[i] = acc;
}

// ---------------------------------------------------------------------------
extern "C" void kernel_launch(void* const* d_in, const int* in_sizes, int n_in,
                              void* d_out, int out_size, void* d_ws, size_t ws_size,
                              hipStream_t stream) {
    const float* h_in   = (const float*)d_in[0];
    const int*   src    = (const int*)  d_in[1];
    const int*   dst    = (const int*)  d_in[2];
    const int*   gid    = (const int*)  d_in[3];
    const float* w_init = (const float*)d_in[4];
    const float* lw[3]    = {(const float*)d_in[5],  (const float*)d_in[9],  (const float*)d_in[13]};
    const float* lb[3]    = {(const float*)d_in[6],  (const float*)d_in[10], (const float*)d_in[14]};
    const float* gamma[3] = {(const float*)d_in[7],  (const float*)d_in[11], (const float*)d_in[15]};
    const float* beta[3]  = {(const float*)d_in[8],  (const float*)d_in[12], (const float*)d_in[16]};
    const float* fc_w = (const float*)d_in[17];
    const float* fc_b = (const float*)d_in[18];
    float* out = (float*)d_out;

    const int nScanBlocks = (N_NODES + SCAN_CHUNK - 1) / SCAN_CHUNK;   // 98

    // ---- workspace layout (all offsets in 4-byte elements, 16B aligned) ----
    char* wsb = (char*)d_ws;
    int*   outc      = (int*)wsb;                                  // N
    int*   inc       = outc + N_NODES;                             // N
    int*   row_start = inc + N_NODES;                              // N+4 (pad)
    int*   cursor    = row_start + N_NODES + 4;                    // N
    int*   blockSums = cursor + N_NODES;                           // 128
    int*   csr_src   = blockSums + 128;                            // E
    float* norm_src  = (float*)(csr_src + N_EDGES);                // N
    float* norm_dst  = norm_src + N_NODES;                         // N
    float* sums      = norm_dst + N_NODES;                         // 256
    float* sumsq     = sums + HID;                                 // 256
    float* hg        = sumsq + HID;                                // 128*256
    float* bufA      = hg + N_GRAPH * HID;                         // N*256
    float* bufB      = bufA + (size_t)N_NODES * HID;               // N*256

    // ---- 1) degrees -> norms; CSR build (static graph, built once/call) ----
    hipMemsetAsync(outc, 0, 2 * (size_t)N_NODES * sizeof(int), stream);
    gcn_degcnt<<<(N_EDGES + 255) / 256, 256, 0, stream>>>(src, dst, outc, inc, N_EDGES);
    gcn_norms<<<(N_NODES + 255) / 256, 256, 0, stream>>>(outc, inc, norm_src, norm_dst, N_NODES);

    gcn_scan1<<<nScanBlocks, 256, 0, stream>>>(inc, row_start, blockSums, N_NODES);
    gcn_scan2<<<1, 64, 0, stream>>>(blockSums, nScanBlocks);
    gcn_scan3<<<(N_NODES + 255) / 256, 256, 0, stream>>>(row_start, blockSums, N_NODES, N_EDGES);
    hipMemcpyAsync(cursor, row_start, N_NODES * sizeof(int),
                   hipMemcpyDeviceToDevice, stream);
    gcn_csr_fill<<<(N_EDGES + 255) / 256, 256, 0, stream>>>(src, dst, cursor, csr_src, N_EDGES);

    // ---- 2) init projection, pre-scaled by norm_src: bufA = diag(ns) h W^T ----
    {
        int ntiles = (N_NODES / 16) * (NC / 16);
        gcn_gemm_wmma_f32<<<(ntiles + 7) / 8, 256, 0, stream>>>(
            h_in, w_init, bufA, norm_src, nullptr, N_NODES, F_IN, NC, 1);
    }

    // ---- 3) three GraphConv + BN + ReLU layers (atomic-free aggregation) ----
    int din = NC;
    for (int i = 0; i < 3; ++i) {
        gcn_csr_agg<<<(N_NODES + 7) / 8, 256, 0, stream>>>(
            bufA, row_start, csr_src, bufB, N_NODES, din);

        int ntiles = (N_NODES / 16) * (HID / 16);
        gcn_gemm_wmma_f32<<<(ntiles + 7) / 8, 256, 0, stream>>>(
            bufB, lw[i], bufA, norm_dst, lb[i], N_NODES, din, HID, 0);

        hipMemsetAsync(sums, 0, 2 * HID * sizeof(float), stream);
        gcn_colstats<<<(N_NODES + 511) / 512, 256, 0, stream>>>(bufA, sums, sumsq, N_NODES, 512);
        gcn_bn_relu<<<(int)(((size_t)N_NODES * HID + 255) / 256), 256, 0, stream>>>(
            bufA, sums, sumsq, gamma[i], beta[i],
            (i < 2) ? norm_src : nullptr, N_NODES);
        din = HID;
    }

    // ---- 4) per-graph max pool + FC ----
    hipMemsetAsync(hg, 0, (size_t)N_GRAPH * HID * sizeof(float), stream);
    gcn_pool_max<<<(int)(((size_t)N_NODES * HID + 255) / 256), 256, 0, stream>>>(
        bufA, gid, (unsigned*)hg, N_NODES);
    gcn_fc<<<(N_GRAPH * OUTC + 255) / 256, 256, 0, stream>>>(hg, fc_w, fc_b, out);
}